// RandomProjectionQuantizer_46540265619576
// MI455X (gfx1250) — compile-verified
//
#include <hip/hip_runtime.h>
#include <hip/hip_bf16.h>

// ---------------------------------------------------------------------------
// RandomProjectionQuantizer, fused for MI455X (gfx1250, wave32).
//
// Roofline: x = 512 MiB fp32 dominates (proj/codebook L2-resident on 192MB L2)
// -> 512 MiB / 23.3 TB/s ~= 22 us floor; ~8.6 GFLOP total => purely HBM-bound.
// Strategy: one block (8 waves) per 16-row tile, K split across waves (512
// k-columns each) for 16384 waves of concurrency; partial 16x16 accumulators
// combined via LDS in a FIXED order (deterministic, no float atomics).
// fp32 WMMA (V_WMMA_F32_16X16X4_F32) for both GEMMs; x loads NT + prefetch.
// ---------------------------------------------------------------------------

typedef float v2f __attribute__((ext_vector_type(2)));
typedef float v4f __attribute__((ext_vector_type(4)));
typedef float v8f __attribute__((ext_vector_type(8)));

#define INPUT_DIM  4096
#define CODE_DIM   16
#define CODE_SIZE  4096
#define ROWS       (8 * 4096)     // B*N flattened
#define WAVES_PER_BLOCK 8
#define K_PER_WAVE (INPUT_DIM / WAVES_PER_BLOCK)   // 512
#define TILES_PER_WAVE (CODE_SIZE / 16 / WAVES_PER_BLOCK)  // 32

// DS_SWIZZLE_B32 group-of-32 XOR patterns: offset = xor_mask<<10 | and_mask
#define SWZ_X1 0x041F
#define SWZ_X2 0x081F
#define SWZ_X4 0x101F
#define SWZ_X8 0x201F

template <int PAT>
__device__ __forceinline__ float ds_swz_f(float v) {
    return __int_as_float(__builtin_amdgcn_ds_swizzle(__float_as_int(v), PAT));
}
template <int PAT>
__device__ __forceinline__ int ds_swz_i(int v) {
    return __builtin_amdgcn_ds_swizzle(v, PAT);
}

// ---- pre-pass: ||codebook[c]||^2 into workspace (16 KB, L2-resident) ------
__global__ void rpq_cnorm2_kernel(const float* __restrict__ cb,
                                  float* __restrict__ cn2) {
    int c = blockIdx.x * blockDim.x + threadIdx.x;
    if (c < CODE_SIZE) {
        const v4f* p = (const v4f*)(cb + (size_t)c * CODE_DIM);
        v4f a = p[0], b = p[1], d = p[2], e = p[3];
        float s = 0.f;
#pragma unroll
        for (int i = 0; i < 4; ++i)
            s += a[i]*a[i] + b[i]*b[i] + d[i]*d[i] + e[i]*e[i];
        cn2[c] = s;
    }
}

// ---- fused projection + normalize + nearest-code search -------------------
// WMMA f32 16x16x4 register contracts (ISA 7.12.2):
//   A (16x4):  lanes 0-15 / 16-31 hold row M=lane&15; VGPR pair = K{0,1} for
//              lanes<16 and K{2,3} for lanes>=16.
//   B (4x16):  VGPR pair holds rows K{0,1}/K{2,3} striped over N=lane&15.
//   C/D:       VGPR j: lanes0-15 -> M=j, lanes16-31 -> M=j+8, N=lane&15.
__global__ __launch_bounds__(256) void rpq_search_kernel(
        const float* __restrict__ x,     // [ROWS, INPUT_DIM]
        const float* __restrict__ P,     // [INPUT_DIM, CODE_DIM]
        const float* __restrict__ cb,    // [CODE_SIZE, CODE_DIM]
        const float* __restrict__ cn2,   // [CODE_SIZE]
        int* __restrict__ out) {         // [ROWS]
    // partials[w][r*16+e]; partials[0..255] is reused as the normalized tile
    __shared__ float partials[WAVES_PER_BLOCK * 256];
    __shared__ float candV[WAVES_PER_BLOCK][16];
    __shared__ int   candI[WAVES_PER_BLOCK][16];

    const int tid     = threadIdx.x;
    const int lane    = tid & 31;
    const int waveBlk = tid >> 5;
    const int rowBase = blockIdx.x * 16;         // one 16-row tile per block
    const int half    = lane >> 4;               // 0: K{0,1}; 1: K{2,3}
    const int m       = lane & 15;

    // ------- Phase 1: partial proj over this wave's K slice ----------------
    v8f acc = {};
    const float* xA = x + (size_t)(rowBase + m) * INPUT_DIM + half * 2;
    const float* pB = P + (size_t)(half * 2) * CODE_DIM + m;
    const int kBase = waveBlk * K_PER_WAVE;
#pragma unroll 1
    for (int k0 = kBase; k0 < kBase + K_PER_WAVE; k0 += 64) {
        // speculative streaming prefetch ~1KB ahead
        __builtin_prefetch(xA + k0 + 256, 0, 0);
        v2f a[16], b[16];
#pragma unroll
        for (int t = 0; t < 16; ++t) {           // distinct regs -> pipelined
            const int k = k0 + 4 * t;
            a[t] = __builtin_nontemporal_load((const v2f*)(xA + k));
            b[t].x = pB[k * CODE_DIM];
            b[t].y = pB[k * CODE_DIM + CODE_DIM];
        }
#pragma unroll
        for (int t = 0; t < 16; ++t)
            acc = __builtin_amdgcn_wmma_f32_16x16x4_f32(
                      false, a[t], false, b[t], (short)0, acc, false, false);
    }

    // ------- Deterministic cross-wave reduction of the 16x16 tile ----------
#pragma unroll
    for (int j = 0; j < 8; ++j)                  // C layout -> [row][e] in LDS
        partials[waveBlk * 256 + (j + half * 8) * 16 + m] = acc[j];
    __syncthreads();

    // thread t owns element (r = t/16, e = t%16); sum waves in fixed order
    float v = partials[tid];
#pragma unroll
    for (int w = 1; w < WAVES_PER_BLOCK; ++w) v += partials[w * 256 + tid];

    // ------- Normalize rows, fold in -2 scale (16-lane groups == rows) -----
    float ss = v * v;
    ss += ds_swz_f<SWZ_X1>(ss);
    ss += ds_swz_f<SWZ_X2>(ss);
    ss += ds_swz_f<SWZ_X4>(ss);
    ss += ds_swz_f<SWZ_X8>(ss);
    float scale = -2.0f / fmaxf(sqrtf(ss), 1e-12f);
    partials[tid] = v * scale;                   // xn * (-2), [row][e]
    __syncthreads();

    // ------- Phase 2: argmin_c ||c||^2 - 2 xn.c over this wave's tiles -----
    v2f a2[4];                                   // xn in A-layout (per wave)
    const float* myRow = &partials[m * 16 + half * 2];
#pragma unroll
    for (int t = 0; t < 4; ++t) a2[t] = *(const v2f*)(myRow + 4 * t);

    float bestv[8];
    int   besti[8];
#pragma unroll
    for (int j = 0; j < 8; ++j) { bestv[j] = 3.402823466e38f; besti[j] = 0; }

    const float* cbB = cb + m * CODE_DIM + half * 2;
    const int tileBeg = waveBlk * TILES_PER_WAVE;
#pragma unroll 1
    for (int tile = tileBeg; tile < tileBeg + TILES_PER_WAVE; ++tile) {
        const int code = tile * 16 + m;          // same for both halves, all j
        const float cn = cn2[code];
        v8f s = { cn, cn, cn, cn, cn, cn, cn, cn };  // seed C with ||c||^2
#pragma unroll
        for (int t = 0; t < 4; ++t) {
            v2f bb = *(const v2f*)(cbB + (size_t)tile * 256 + 4 * t);
            s = __builtin_amdgcn_wmma_f32_16x16x4_f32(
                    false, a2[t], false, bb, (short)0, s, false, false);
        }
#pragma unroll
        for (int j = 0; j < 8; ++j) {
            if (s[j] < bestv[j]) { bestv[j] = s[j]; besti[j] = code; }
        }
    }

    // ------- In-wave 16-lane min+argmin reduction --------------------------
#define ARGMIN_STEP(PAT)                                                     \
    {                                                                        \
        float ov = ds_swz_f<PAT>(bestv[j]);                                  \
        int   oi = ds_swz_i<PAT>(besti[j]);                                  \
        if (ov < bestv[j] || (ov == bestv[j] && oi < besti[j])) {            \
            bestv[j] = ov; besti[j] = oi;  /* first-occurrence tie-break */  \
        }                                                                    \
    }
#pragma unroll
    for (int j = 0; j < 8; ++j) {
        ARGMIN_STEP(SWZ_X1)
        ARGMIN_STEP(SWZ_X2)
        ARGMIN_STEP(SWZ_X4)
        ARGMIN_STEP(SWZ_X8)
    }
#undef ARGMIN_STEP

    if (m == 0) {                                // lanes 0 & 16 publish 8 rows
#pragma unroll
        for (int j = 0; j < 8; ++j) {
            candV[waveBlk][half * 8 + j] = bestv[j];
            candI[waveBlk][half * 8 + j] = besti[j];
        }
    }
    __syncthreads();

    // ------- Final cross-wave scan (wave order == ascending code ranges) ---
    if (tid < 16) {
        float bv = candV[0][tid];
        int   bi = candI[0][tid];
#pragma unroll
        for (int w = 1; w < WAVES_PER_BLOCK; ++w) {
            float ov = candV[w][tid];
            int   oi = candI[w][tid];
            if (ov < bv || (ov == bv && oi < bi)) { bv = ov; bi = oi; }
        }
        out[rowBase + tid] = bi;
    }
}

extern "C" void kernel_launch(void* const* d_in, const int* in_sizes, int n_in,
                              void* d_out, int out_size, void* d_ws, size_t ws_size,
                              hipStream_t stream) {
    (void)in_sizes; (void)n_in; (void)out_size; (void)ws_size;
    const float* x  = (const float*)d_in[0];   // [8,4096,4096] f32
    const float* P  = (const float*)d_in[1];   // [4096,16]     f32
    const float* cb = (const float*)d_in[2];   // [4096,16]     f32
    float* cn2 = (float*)d_ws;                 // 4096 f32 scratch
    int*   out = (int*)d_out;                  // [8,4096] int32 indices

    rpq_cnorm2_kernel<<<CODE_SIZE / 256, 256, 0, stream>>>(cb, cn2);
    rpq_search_kernel<<<ROWS / 16, 256, 0, stream>>>(x, P, cb, cn2, out);
}